// DynamicDecoder_77421080477747
// MI455X (gfx1250) — compile-verified
//
#include <hip/hip_runtime.h>
#include <hip/hip_bf16.h>
#include <math.h>

#define H 256
#define TWOH 512
#define POOL 5
#define BATCH 64
#define MSEQ 600
#define ROWS (BATCH * MSEQ) /* 38400 */
#define MAXIT 4

typedef __attribute__((ext_vector_type(2))) float v2f;
typedef __attribute__((ext_vector_type(8))) float v8f;

// ---------------------------------------------------------------- init
__global__ __launch_bounds__(256) void init_kernel(float* hidden, int* start) {
    int i = blockIdx.x * 256 + threadIdx.x;
    if (i < BATCH * H) hidden[i] = 0.0f;
    if (i < BATCH) start[i] = 0;
}

// ------------------------------------------------- r = [us,us,h] @ mlp_w^T + b
__global__ __launch_bounds__(256) void r_kernel(const float* __restrict__ info,
                                                const int* __restrict__ start,
                                                const float* __restrict__ hidden,
                                                const float* __restrict__ W,
                                                const float* __restrict__ b,
                                                float* __restrict__ r) {
    int bb = blockIdx.x, o = threadIdx.x;
    const float* us = info + ((size_t)bb * MSEQ + start[bb]) * TWOH;
    const float* w = W + (size_t)o * (5 * H);
    float acc = b[o];
    for (int k = 0; k < TWOH; ++k) acc += us[k] * (w[k] + w[TWOH + k]);
    const float* h = hidden + bb * H;
    for (int k = 0; k < H; ++k) acc += h[k] * w[2 * TWOH + k];
    r[bb * H + o] = acc;
}

// ------------------------- rW[b,n] = m1_b[n] + sum_o r[b,o]*m1_w[n, 512+o]
__global__ __launch_bounds__(256) void rw_kernel(const float* __restrict__ r,
                                                 const float* __restrict__ m1w,
                                                 const float* __restrict__ m1b,
                                                 float* __restrict__ rW) {
    int bb = blockIdx.y;
    int n = blockIdx.x * 256 + threadIdx.x; // 0..1279
    const float* w = m1w + (size_t)n * (3 * H) + TWOH;
    const float* rr = r + bb * H;
    float acc = m1b[n];
    for (int o = 0; o < H; ++o) acc += rr[o] * w[o];
    rW[(size_t)bb * (POOL * H) + n] = acc;
}

// ---------------------------------------------------------------------------
// Ypre[m, n] = sum_k X[m,k] * W[n,k] + addvec[batch(m)*addStride + n]
// out[m, g]  = max_{p<5} Ypre[m, g*5+p]
//
// Block tile: M=128 rows, N=160 cols (=32 maxout groups). 8 waves:
//   wave = {mt(0..3): 32-row subtile, nh(0..1): 80-col half} -> 2x5 frags/wave.
// LDS tiles stored as K-pair float2 so each WMMA A/B fragment is a single
// aligned ds_load_b64 (no cross-row gathers, no repacking movs).
// Epilogue: two 64-row phases through the same LDS buffer (pool-5 maxout).
// ---------------------------------------------------------------------------
#define A_LDP 17   /* pairs per A row (16 + 1 pad, odd -> bank spread) */
#define B_LDP 176  /* cols per B pair-row (160 + 16 pad; 2*176 % 64 == 32) */

__global__ __launch_bounds__(256) void gemm_maxout(const float* __restrict__ X,
                                                   int ldX, int K,
                                                   const float* __restrict__ W,
                                                   int ldW,
                                                   const float* __restrict__ addvec,
                                                   int addStride,
                                                   float* __restrict__ out) {
    __shared__ __align__(16) float smem[64 * 161]; // 41.2 KB
    v2f* As2 = (v2f*)smem;            // [128][A_LDP] pairs  (2176 f2)
    v2f* Bs2 = As2 + 128 * A_LDP;     // [16][B_LDP]  pairs  (2816 f2)
    float* Ys = smem;                 // [64][161] (aliased after K-loop)

    const int tid = threadIdx.x;
    const int lane = tid & 31;
    const int wave = tid >> 5;
    const int mt = wave & 3;       // 32-row subtile
    const int nh = wave >> 2;      // 80-col half
    const int half = lane >> 4;    // 0: lanes 0-15, 1: lanes 16-31
    const int l16 = lane & 15;
    const int rowBase = blockIdx.y * 128;
    const int nBase = blockIdx.x * 160;

    v8f acc[2][5];
    for (int mi = 0; mi < 2; ++mi)
        for (int f = 0; f < 5; ++f)
            for (int i = 0; i < 8; ++i) acc[mi][f][i] = 0.0f;

    for (int k0 = 0; k0 < K; k0 += 32) {
        __syncthreads();
        // stage A tile 128x32 as 16 K-pairs per row (coalesced b64 reads)
        for (int e = tid; e < 128 * 16; e += 256) {
            int rr = e >> 4, j = e & 15;
            As2[rr * A_LDP + j] =
                *(const v2f*)&X[(size_t)(rowBase + rr) * ldX + k0 + 2 * j];
        }
        // stage B tile 32x160: Bs2[j][n] = (W[n][k0+2j], W[n][k0+2j+1])
        for (int e = tid; e < 160 * 16; e += 256) {
            int nn = e >> 4, j = e & 15;
            Bs2[j * B_LDP + nn] =
                *(const v2f*)&W[(size_t)(nBase + nn) * ldW + k0 + 2 * j];
        }
        if (k0 + 32 < K) { // prefetch next stage into L2/L0
            __builtin_prefetch(
                &X[(size_t)(rowBase + (tid >> 1)) * ldX + k0 + 32 + (tid & 1) * 16], 0, 0);
            __builtin_prefetch(
                &W[(size_t)(nBase + (tid % 160)) * ldW + k0 + 32], 0, 0);
        }
        __syncthreads();
        for (int kk = 0; kk < 32; kk += 4) {
            int j = (kk >> 1) + half; // even K index ka=kk+2*half -> pair j
            v2f a0 = As2[(mt * 32 + l16) * A_LDP + j];
            v2f a1 = As2[(mt * 32 + 16 + l16) * A_LDP + j];
            for (int f = 0; f < 5; ++f) {
                v2f bf = Bs2[j * B_LDP + nh * 80 + f * 16 + l16];
                acc[0][f] = __builtin_amdgcn_wmma_f32_16x16x4_f32(
                    false, a0, false, bf, (short)0, acc[0][f], false, false);
                acc[1][f] = __builtin_amdgcn_wmma_f32_16x16x4_f32(
                    false, a1, false, bf, (short)0, acc[1][f], false, false);
            }
        }
    }
    __syncthreads(); // tiles dead; smem becomes Ys
    // epilogue in two 64-row phases: +addvec, stage, maxout 5->1, store
    for (int phase = 0; phase < 2; ++phase) {
        if ((mt >> 1) == phase) {
            for (int mi = 0; mi < 2; ++mi) {
                for (int f = 0; f < 5; ++f) {
                    int colL = nh * 80 + f * 16 + l16;
                    int colG = nBase + colL;
                    for (int rci = 0; rci < 8; ++rci) {
                        int rowL = mt * 32 + mi * 16 + rci + half * 8;
                        int batch = (rowBase + rowL) / MSEQ;
                        Ys[(rowL - phase * 64) * 161 + colL] =
                            acc[mi][f][rci] + addvec[(size_t)batch * addStride + colG];
                    }
                }
            }
        }
        __syncthreads();
        for (int e = tid; e < 64 * 32; e += 256) {
            int rr = e >> 5, g = e & 31;
            const float* y = &Ys[rr * 161 + g * 5];
            float v = y[0];
            for (int p = 1; p < POOL; ++p) v = fmaxf(v, y[p]);
            out[(size_t)(rowBase + phase * 64 + rr) * H + blockIdx.x * 32 + g] = v;
        }
        __syncthreads();
    }
}

// ------------------------- m3: scores[row] = max_p( [m1,m2] . w3[p] + b3[p] )
__global__ __launch_bounds__(256) void m3_scores(const float* __restrict__ m1o,
                                                 const float* __restrict__ m2o,
                                                 const float* __restrict__ w3,
                                                 const float* __restrict__ b3,
                                                 float* __restrict__ scores) {
    int row = blockIdx.x * 8 + (threadIdx.x >> 5);
    int lane = threadIdx.x & 31;
    const float* x1 = m1o + (size_t)row * H;
    const float* x2 = m2o + (size_t)row * H;
    float a0 = 0, a1 = 0, a2 = 0, a3 = 0, a4 = 0;
    for (int k = lane; k < TWOH; k += 32) {
        float x = (k < H) ? x1[k] : x2[k - H];
        a0 += x * w3[0 * TWOH + k];
        a1 += x * w3[1 * TWOH + k];
        a2 += x * w3[2 * TWOH + k];
        a3 += x * w3[3 * TWOH + k];
        a4 += x * w3[4 * TWOH + k];
    }
    for (int off = 16; off > 0; off >>= 1) {
        a0 += __shfl_xor(a0, off, 32);
        a1 += __shfl_xor(a1, off, 32);
        a2 += __shfl_xor(a2, off, 32);
        a3 += __shfl_xor(a3, off, 32);
        a4 += __shfl_xor(a4, off, 32);
    }
    if (lane == 0) {
        float v = a0 + b3[0];
        v = fmaxf(v, a1 + b3[1]);
        v = fmaxf(v, a2 + b3[2]);
        v = fmaxf(v, a3 + b3[3]);
        v = fmaxf(v, a4 + b3[4]);
        scores[row] = v;
    }
}

// ------------------------------------- per-batch argmax (first-max tie-break)
__global__ __launch_bounds__(256) void argmax2(const float* __restrict__ ss,
                                               const float* __restrict__ se,
                                               int* __restrict__ ns,
                                               int* __restrict__ ne,
                                               int writeFinal,
                                               float* __restrict__ outSE) {
    __shared__ float sv[256];
    __shared__ int si[256];
    int bb = blockIdx.x, t = threadIdx.x;
    for (int which = 0; which < 2; ++which) {
        const float* s = (which == 0 ? ss : se) + (size_t)bb * MSEQ;
        float best = -__builtin_inff();
        int bi = 0;
        for (int i = t; i < MSEQ; i += 256) {
            float v = s[i];
            if (v > best) { best = v; bi = i; }
        }
        sv[t] = best; si[t] = bi;
        __syncthreads();
        for (int o = 128; o > 0; o >>= 1) {
            if (t < o) {
                float v2 = sv[t + o]; int i2 = si[t + o];
                if (v2 > sv[t] || (v2 == sv[t] && i2 < si[t])) { sv[t] = v2; si[t] = i2; }
            }
            __syncthreads();
        }
        if (t == 0) {
            int idx = si[0];
            if (which == 0) { ns[bb] = idx; if (writeFinal) outSE[bb] = (float)idx; }
            else            { ne[bb] = idx; if (writeFinal) outSE[BATCH + bb] = (float)idx; }
        }
        __syncthreads();
    }
}

// --------------------------------------------------- GRU cell, block per batch
__global__ __launch_bounds__(256) void gru_kernel(const float* __restrict__ info,
                                                  const int* __restrict__ ns,
                                                  float* __restrict__ hidden,
                                                  const float* __restrict__ wih,
                                                  const float* __restrict__ whh,
                                                  const float* __restrict__ bih,
                                                  const float* __restrict__ bhh) {
    __shared__ float hold[H];
    __shared__ float GI[3 * H];
    __shared__ float GH[3 * H];
    int bb = blockIdx.x, t = threadIdx.x;
    hold[t] = hidden[bb * H + t];
    __syncthreads();
    const float* us = info + ((size_t)bb * MSEQ + ns[bb]) * TWOH;
    for (int j = t; j < 3 * H; j += 256) {
        const float* wi = wih + (size_t)j * (4 * H);
        float gi = bih[j];
        for (int k = 0; k < TWOH; ++k) gi += us[k] * (wi[k] + wi[TWOH + k]);
        const float* wh = whh + (size_t)j * H;
        float gh = bhh[j];
        for (int k = 0; k < H; ++k) gh += hold[k] * wh[k];
        GI[j] = gi; GH[j] = gh;
    }
    __syncthreads();
    {
        int i = t;
        float r = 1.0f / (1.0f + __expf(-(GI[i] + GH[i])));
        float z = 1.0f / (1.0f + __expf(-(GI[H + i] + GH[H + i])));
        float n = tanhf(GI[2 * H + i] + r * GH[2 * H + i]);
        hidden[bb * H + i] = (1.0f - z) * n + z * hold[i];
    }
}

// ---------------------------------------------------------------------------
extern "C" void kernel_launch(void* const* d_in, const int* in_sizes, int n_in,
                              void* d_out, int out_size, void* d_ws, size_t ws_size,
                              hipStream_t stream) {
    const float* info     = (const float*)d_in[0];
    const float* gru_bhh  = (const float*)d_in[1];
    const float* gru_bih  = (const float*)d_in[2];
    const float* gru_whh  = (const float*)d_in[3];
    const float* gru_wih  = (const float*)d_in[4];
    const float* he_m1_b  = (const float*)d_in[5];
    const float* he_m1_w  = (const float*)d_in[6];
    const float* he_m2_b  = (const float*)d_in[7];
    const float* he_m2_w  = (const float*)d_in[8];
    const float* he_m3_b  = (const float*)d_in[9];
    const float* he_m3_w  = (const float*)d_in[10];
    const float* he_mlp_b = (const float*)d_in[11];
    const float* he_mlp_w = (const float*)d_in[12];
    const float* hs_m1_b  = (const float*)d_in[13];
    const float* hs_m1_w  = (const float*)d_in[14];
    const float* hs_m2_b  = (const float*)d_in[15];
    const float* hs_m2_w  = (const float*)d_in[16];
    const float* hs_m3_b  = (const float*)d_in[17];
    const float* hs_m3_w  = (const float*)d_in[18];
    const float* hs_mlp_b = (const float*)d_in[19];
    const float* hs_mlp_w = (const float*)d_in[20];

    float* out = (float*)d_out; // [start(64)] [end(64)] [scores 4*2*64*600]

    float* ws = (float*)d_ws;
    float* hidden = ws;              ws += BATCH * H;
    float* rbuf   = ws;              ws += BATCH * H;
    float* rW     = ws;              ws += BATCH * POOL * H;
    float* out1   = ws;              ws += (size_t)ROWS * H;
    float* out2   = ws;              ws += (size_t)ROWS * H;
    int* startA = (int*)ws;
    int* startB = startA + BATCH;
    int* endB   = startB + BATCH;

    init_kernel<<<dim3((BATCH * H + 255) / 256), 256, 0, stream>>>(hidden, startA);

    int* curStart = startA;
    int* nxtStart = startB;
    for (int it = 0; it < MAXIT; ++it) {
        for (int se = 0; se < 2; ++se) {
            const float* mlp_w = se ? he_mlp_w : hs_mlp_w;
            const float* mlp_b = se ? he_mlp_b : hs_mlp_b;
            const float* m1_w  = se ? he_m1_w  : hs_m1_w;
            const float* m1_b  = se ? he_m1_b  : hs_m1_b;
            const float* m2_w  = se ? he_m2_w  : hs_m2_w;
            const float* m2_b  = se ? he_m2_b  : hs_m2_b;
            const float* m3_w  = se ? he_m3_w  : hs_m3_w;
            const float* m3_b  = se ? he_m3_b  : hs_m3_b;
            float* scores = out + 2 * BATCH + (size_t)(it * 2 + se) * ROWS;

            r_kernel<<<BATCH, 256, 0, stream>>>(info, curStart, hidden, mlp_w, mlp_b, rbuf);
            rw_kernel<<<dim3(POOL, BATCH), 256, 0, stream>>>(rbuf, m1_w, m1_b, rW);
            // m1: info-part of r_u GEMM (K=512), r-part folded into rW
            gemm_maxout<<<dim3(8, ROWS / 128), 256, 0, stream>>>(info, TWOH, TWOH,
                                                                 m1_w, 3 * H,
                                                                 rW, POOL * H, out1);
            // m2: K=256, plain bias (batch stride 0)
            gemm_maxout<<<dim3(8, ROWS / 128), 256, 0, stream>>>(out1, H, H,
                                                                 m2_w, H,
                                                                 m2_b, 0, out2);
            m3_scores<<<ROWS / 8, 256, 0, stream>>>(out1, out2, m3_w, m3_b, scores);
        }
        argmax2<<<BATCH, 256, 0, stream>>>(out + 2 * BATCH + (size_t)(it * 2) * ROWS,
                                           out + 2 * BATCH + (size_t)(it * 2 + 1) * ROWS,
                                           nxtStart, endB, (it == MAXIT - 1) ? 1 : 0, out);
        gru_kernel<<<BATCH, 256, 0, stream>>>(info, nxtStart, hidden,
                                              gru_wih, gru_whh, gru_bih, gru_bhh);
        int* tmp = curStart; curStart = nxtStart; nxtStart = tmp;
    }
}